// VQVAE_65034394796676
// MI455X (gfx1250) — compile-verified
//
#include <hip/hip_runtime.h>

// VQ-VAE nearest-codebook quantization for MI455X (gfx1250, wave32).
//
// Shapes: cw_q (B=1024, 512) f32, codebook (C=128, K=1024, E=4) f32.
// Outputs (concatenated in d_out): cw_embed (1024,512) f32, one_hot (1024,128,1024) f32.
//
// Kernel 1: per-wave 16-batch-row x 1-channel tile; V_WMMA_F32_16X16X4_F32
//           computes xb = x . codebook^T (contraction length E=4 == WMMA K=4).
//           argmin over k uses score = b_sq[k] - 2*xb (x_sq is row-constant).
//           b_sq computed lane-locally from a b128 codebook load (no LDS
//           bpermute in the hot loop); final cross-lane argmin is a
//           branch-free u64-key min reduction.
// Kernel 2: streaming one-hot fill, 512 MB of non-temporal b128 stores
//           (the entire runtime cost: ~22us at 23.3 TB/s).

typedef float v2f __attribute__((ext_vector_type(2)));
typedef float v4f __attribute__((ext_vector_type(4)));
typedef float v8f __attribute__((ext_vector_type(8)));

#define B_DIM 1024
#define C_DIM 128
#define K_DIM 1024
#define E_DIM 4
#define CW    512   // C_DIM * E_DIM

// Order-preserving float -> uint32 map (f0 < f1  <=>  key(f0) < key(f1)).
static __device__ __forceinline__ unsigned int fkey(float f) {
    unsigned int u = __float_as_uint(f);
    return u ^ ((unsigned int)((int)u >> 31) | 0x80000000u);
}

__global__ __launch_bounds__(256) void vq_argmin_wmma_kernel(
    const float* __restrict__ cw_q,      // (B, 512)
    const float* __restrict__ codebook,  // (C, K, 4)
    float* __restrict__ embed_out,       // (B, 512)
    int*   __restrict__ idx_out)         // (B, C) workspace
{
    const int wave = (blockIdx.x * blockDim.x + threadIdx.x) >> 5;
    const int lane = threadIdx.x & 31;
    const int c    = wave & (C_DIM - 1);
    const int b0   = (wave / C_DIM) << 4;   // 16-row batch tile
    const int half = lane >> 4;             // 0: K-elems {0,1}; 1: {2,3}
    const int nl   = lane & 15;             // row (A) / column (B) within tile
    const int e0   = half * 2;

    // A tile (16x4 f32): lane holds row m=nl, contraction elems e0,e0+1.
    v2f a = *(const v2f*)(cw_q + (size_t)(b0 + nl) * CW + c * E_DIM + e0);

    const float* cb = codebook + (size_t)c * K_DIM * E_DIM;

    float minv[8];
    int   mini[8];
#pragma unroll
    for (int g = 0; g < 8; ++g) { minv[g] = 3.4e38f; mini[g] = 0; }

#pragma unroll 2
    for (int kt = 0; kt < K_DIM / 16; ++kt) {
        const int k0 = kt * 16;
        // Full codebook entry k0+nl (16B aligned -> global_load_b128).
        v4f cv = *(const v4f*)(cb + (size_t)(k0 + nl) * E_DIM);

        // ||codebook[k]||^2 computed lane-locally: no cross-lane traffic.
        float bsq = cv.x * cv.x + cv.y * cv.y + cv.z * cv.z + cv.w * cv.w;

        // B tile (4x16 f32): lane holds column n=nl, contraction elems e0,e0+1.
        v2f bv;
        bv.x = half ? cv.z : cv.x;
        bv.y = half ? cv.w : cv.y;

        v8f acc = {};
        acc = __builtin_amdgcn_wmma_f32_16x16x4_f32(
            /*neg_a=*/false, a, /*neg_b=*/false, bv,
            /*c_mod=*/(short)0, acc, /*reuse_a=*/false, /*reuse_b=*/false);

        const int cand = k0 + nl;  // this lane's candidate k (column n=nl)
#pragma unroll
        for (int g = 0; g < 8; ++g) {
            // D[g] on this lane = xb for batch row (g + 8*half), column nl.
            float score = __builtin_fmaf(-2.0f, acc[g], bsq);
            if (score < minv[g]) { minv[g] = score; mini[g] = cand; }
        }
    }

    // Branch-free argmin reduction across the 16 lanes of each half.
    // Key = (order-preserving score bits) << 32 | k : u64 min gives the
    // smallest score, ties broken to the smallest k (jnp.argmin semantics).
    unsigned long long key[8];
#pragma unroll
    for (int g = 0; g < 8; ++g)
        key[g] = ((unsigned long long)fkey(minv[g]) << 32) | (unsigned int)mini[g];

#pragma unroll
    for (int off = 8; off >= 1; off >>= 1) {
#pragma unroll
        for (int g = 0; g < 8; ++g) {
            unsigned long long o = __shfl_xor(key[g], off, 16);
            key[g] = (o < key[g]) ? o : key[g];
        }
    }

    // One lane per row writes the index + gathers the 4-float embedding.
#pragma unroll
    for (int g = 0; g < 8; ++g) {
        if (nl == g) {
            const int b  = b0 + g + 8 * half;
            const int id = (int)(key[g] & 0xFFFFFFFFu);
            idx_out[(size_t)b * C_DIM + c] = id;
            v4f e = *(const v4f*)(cb + (size_t)id * E_DIM);
            *(v4f*)(embed_out + (size_t)b * CW + c * E_DIM) = e;
        }
    }
}

// One block per (b,c) row: 256 threads x float4 == K=1024 floats exactly.
// Pure streaming writes -> non-temporal hint (don't pollute L2 with 512 MB).
__global__ __launch_bounds__(256) void vq_onehot_kernel(
    const int* __restrict__ idx_in,   // (B*C)
    float*     __restrict__ onehot)   // (B*C, K)
{
    const int row = blockIdx.x;
    const int idx = idx_in[row];      // uniform per block -> scalar load
    const int k0  = threadIdx.x * 4;
    v4f v;
    v.x = (k0 + 0 == idx) ? 1.0f : 0.0f;
    v.y = (k0 + 1 == idx) ? 1.0f : 0.0f;
    v.z = (k0 + 2 == idx) ? 1.0f : 0.0f;
    v.w = (k0 + 3 == idx) ? 1.0f : 0.0f;
    __builtin_nontemporal_store(v, (v4f*)(onehot + (size_t)row * K_DIM + k0));
}

extern "C" void kernel_launch(void* const* d_in, const int* in_sizes, int n_in,
                              void* d_out, int out_size, void* d_ws, size_t ws_size,
                              hipStream_t stream) {
    const float* cw_q     = (const float*)d_in[0];
    const float* codebook = (const float*)d_in[1];

    float* out    = (float*)d_out;
    float* embed  = out;                                  // B*CW floats
    float* onehot = out + (size_t)B_DIM * CW;             // B*C*K floats
    int*   idx_ws = (int*)d_ws;                           // B*C ints (512 KB)

    // 8192 (batch-tile, channel) waves, 8 waves (256 threads) per block.
    const int waves = (B_DIM / 16) * C_DIM;
    vq_argmin_wmma_kernel<<<waves / 8, 256, 0, stream>>>(cw_q, codebook, embed, idx_ws);

    // 131072 rows, one block each.
    vq_onehot_kernel<<<B_DIM * C_DIM, 256, 0, stream>>>(idx_ws, onehot);
}